// DKNN_33354716021579
// MI455X (gfx1250) — compile-verified
//
#include <hip/hip_runtime.h>
#include <math.h>

typedef __attribute__((ext_vector_type(2))) float v2f;
typedef __attribute__((ext_vector_type(8))) float v8f;

#define QN 64
#define NN 1024
#define DD 512
#define KTOP 16

// ---------------------------------------------------------------------------
// Kernel 1: squared norms of query rows (64) and neighbor rows (1024).
// ---------------------------------------------------------------------------
__global__ void dknn_norms(const float* __restrict__ query,
                           const float* __restrict__ neighbors,
                           float* __restrict__ qnorm,
                           float* __restrict__ nnorm) {
  int i = blockIdx.x * blockDim.x + threadIdx.x;
  if (i < QN) {
    const float* p = query + (size_t)i * DD;
    float s = 0.f;
    for (int d = 0; d < DD; d += 4) {
      float4 v = *(const float4*)(p + d);
      s += v.x * v.x + v.y * v.y + v.z * v.z + v.w * v.w;
    }
    qnorm[i] = s;
  } else if (i < QN + NN) {
    const float* p = neighbors + (size_t)(i - QN) * DD;
    float s = 0.f;
    for (int d = 0; d < DD; d += 4) {
      float4 v = *(const float4*)(p + d);
      s += v.x * v.x + v.y * v.y + v.z * v.z + v.w * v.w;
    }
    nnorm[i - QN] = s;
  }
}

// ---------------------------------------------------------------------------
// Kernel 2: scores[q,n] = 2*(query . neighbor) - |q|^2 - |n|^2 via full-fp32
// WMMA (V_WMMA_F32_16X16X4_F32). One wave (32 lanes) per 16x16 output tile.
//
// f32 16x4 A layout (per ISA 7.12.2): lanes 0-15 = M rows, VGPR0 holds K=0
// (lanes 0-15) / K=2 (lanes 16-31), VGPR1 holds K=1 / K=3. B (4x16) mirrors
// this with N in place of M. C/D: VGPR r, lanes 0-15 -> (M=r, N=lane),
// lanes 16-31 -> (M=r+8, N=lane-16).
// ---------------------------------------------------------------------------
__global__ void __launch_bounds__(32)
dknn_scores_wmma(const float* __restrict__ query,
                 const float* __restrict__ neighbors,
                 const float* __restrict__ qnorm,
                 const float* __restrict__ nnorm,
                 float* __restrict__ scores) {
  const int tile = blockIdx.x;          // 0 .. (4*64 - 1)
  const int qt   = tile >> 6;           // query tile (64/16 = 4)
  const int nt   = tile & 63;           // neighbor tile (1024/16 = 64)
  const int lane = threadIdx.x;
  const int half = lane >> 4;           // 0: K+0/1, 1: K+2/3
  const int l15  = lane & 15;

  const float* __restrict__ arow = query     + (size_t)(qt * 16 + l15) * DD;
  const float* __restrict__ brow = neighbors + (size_t)(nt * 16 + l15) * DD;

  v8f c = {};
  for (int k = 0; k < DD; k += 4) {
    const int ka = k + half * 2;
    v2f a, b;
    a.x = arow[ka];
    a.y = arow[ka + 1];
    b.x = brow[ka];
    b.y = brow[ka + 1];
    c = __builtin_amdgcn_wmma_f32_16x16x4_f32(
        /*neg_a=*/false, a, /*neg_b=*/false, b,
        /*c_mod=*/(short)0, c, /*reuse_a=*/false, /*reuse_b=*/false);
  }

  const int   nGlob = nt * 16 + l15;
  const float nn    = nnorm[nGlob];
#pragma unroll
  for (int r = 0; r < 8; ++r) {
    const int mGlob = qt * 16 + r + half * 8;
    scores[(size_t)mGlob * NN + nGlob] = 2.0f * c[r] - qnorm[mGlob] - nn;
  }
}

// ---------------------------------------------------------------------------
// Kernel 3: per query row q (one block each):
//   B[n]  = sum_m |s[n] - s[m]|                  (scores row LDS-resident)
//   out[n] = sum_{k=0..15} softmax_n(coef_k * s[n] - B[n])
// coef_k = N + 1 - 2*(k+1).
// ---------------------------------------------------------------------------
__global__ void __launch_bounds__(256)
dknn_softtopk(const float* __restrict__ scores, float* __restrict__ out) {
  __shared__ float s_row[NN];
  __shared__ float b_row[NN];
  __shared__ float red[256];

  const int q   = blockIdx.x;
  const int tid = threadIdx.x;
  constexpr int PER = NN / 256;   // 4 n-values per thread

#pragma unroll
  for (int j = 0; j < PER; ++j)
    s_row[tid + j * 256] = scores[(size_t)q * NN + tid + j * 256];
  __syncthreads();

  // O(N^2) row sum of absolute score differences, entirely from LDS.
#pragma unroll
  for (int j = 0; j < PER; ++j) {
    const float sn = s_row[tid + j * 256];
    float acc = 0.f;
    for (int m = 0; m < NN; ++m)
      acc += fabsf(sn - s_row[m]);
    b_row[tid + j * 256] = acc;
  }
  __syncthreads();

  float outacc[PER];
#pragma unroll
  for (int j = 0; j < PER; ++j) outacc[j] = 0.f;

  for (int k = 0; k < KTOP; ++k) {
    const float coef = (float)(NN + 1 - 2 * (k + 1));

    float lg[PER];
    float lmax = -INFINITY;
#pragma unroll
    for (int j = 0; j < PER; ++j) {
      const int n = tid + j * 256;
      lg[j] = coef * s_row[n] - b_row[n];
      lmax  = fmaxf(lmax, lg[j]);
    }

    // block max reduction
    red[tid] = lmax;
    __syncthreads();
    for (int stride = 128; stride > 0; stride >>= 1) {
      if (tid < stride) red[tid] = fmaxf(red[tid], red[tid + stride]);
      __syncthreads();
    }
    const float gmax = red[0];
    __syncthreads();

    float ex[PER];
    float lsum = 0.f;
#pragma unroll
    for (int j = 0; j < PER; ++j) {
      ex[j] = expf(lg[j] - gmax);
      lsum += ex[j];
    }

    // block sum reduction
    red[tid] = lsum;
    __syncthreads();
    for (int stride = 128; stride > 0; stride >>= 1) {
      if (tid < stride) red[tid] += red[tid + stride];
      __syncthreads();
    }
    const float inv = 1.0f / red[0];
    __syncthreads();

#pragma unroll
    for (int j = 0; j < PER; ++j)
      outacc[j] += ex[j] * inv;
  }

#pragma unroll
  for (int j = 0; j < PER; ++j)
    out[(size_t)q * NN + tid + j * 256] = outacc[j];
}

// ---------------------------------------------------------------------------
// Launch: d_in[0] = query [64,512] f32, d_in[1] = neighbors [1024,512] f32,
// d_out = [64,1024] f32. Workspace: scores (64K f32) + qnorm(64) + nnorm(1024).
// ---------------------------------------------------------------------------
extern "C" void kernel_launch(void* const* d_in, const int* in_sizes, int n_in,
                              void* d_out, int out_size, void* d_ws, size_t ws_size,
                              hipStream_t stream) {
  (void)in_sizes; (void)n_in; (void)out_size; (void)ws_size;
  const float* query     = (const float*)d_in[0];
  const float* neighbors = (const float*)d_in[1];
  float*       out       = (float*)d_out;

  float* scores = (float*)d_ws;          // 64*1024 floats
  float* qnorm  = scores + QN * NN;      // 64 floats
  float* nnorm  = qnorm + QN;            // 1024 floats

  dknn_norms<<<(QN + NN + 255) / 256, 256, 0, stream>>>(query, neighbors, qnorm, nnorm);
  dknn_scores_wmma<<<(QN / 16) * (NN / 16), 32, 0, stream>>>(query, neighbors,
                                                             qnorm, nnorm, scores);
  dknn_softtopk<<<QN, 256, 0, stream>>>(scores, out);
}